// GPNN_CAD_2705829397261
// MI455X (gfx1250) — compile-verified
//
#include <hip/hip_runtime.h>
#include <math.h>

// Shapes
#define B_     8
#define N_     256
#define CE_    64
#define D_     64
#define NN_    (N_ * N_)
#define S_CLS_ 10
#define A_CLS_ 12

typedef float v2f __attribute__((ext_vector_type(2)));
typedef float v8f __attribute__((ext_vector_type(8)));

// Fast transcendentals: v_exp_f32 / v_rcp_f32 hardware paths.
__device__ __forceinline__ float sigm(float x) {
    return __builtin_amdgcn_rcpf(1.f + __expf(-x));
}
__device__ __forceinline__ float ftanh(float x) {
    // tanh(x) = 1 - 2/(exp(2x)+1); saturates correctly at +/-inf.
    return 1.f - 2.f * __builtin_amdgcn_rcpf(1.f + __expf(2.f * x));
}

// f32 WMMA: D(16x16) += A(16x4) * B(4x16).  Exact fp32 (RNE per FMA).
__device__ __forceinline__ v8f wmma4(v2f a, v2f b, v8f c) {
    return __builtin_amdgcn_wmma_f32_16x16x4_f32(false, a, false, b, (short)0, c,
                                                 false, false);
}

// ---------------------------------------------------------------------------
// T1[b,n,m] = Wm[m,0:64] @ H[b,n] + bm[m];  T2[b,n,m] = Wm[m,64:128] @ H[b,n]
// H[b,n,d] = NF[b,d,n].  grid = B*N blocks, 64 threads.
// ---------------------------------------------------------------------------
__global__ void precomp_T(const float* __restrict__ NF, const float* __restrict__ Wm,
                          const float* __restrict__ bm, float* __restrict__ T1,
                          float* __restrict__ T2) {
    int bn = blockIdx.x, b = bn >> 8, n = bn & 255;
    __shared__ float sh[64];
    int m = threadIdx.x;
    sh[m] = NF[(b * 64 + m) * 256 + n];
    __syncthreads();
    float t1 = bm[m], t2 = 0.f;
    const float* wr = Wm + m * 192;
    for (int d = 0; d < 64; ++d) {
        float x = sh[d];
        t1 += wr[d] * x;
        t2 += wr[64 + d] * x;
    }
    T1[bn * 64 + m] = t1;
    T2[bn * 64 + m] = t2;
}

// ---------------------------------------------------------------------------
// M[b,i,w,m] = Wm3[m,:] @ E[b,:,i,w] + T1[b,i,m] + T2[b,w,m]
// One wave per 32-w tile (two 16-col WMMA tiles share each A fragment);
// 4 k-tiles of 16 rows; K=64 via 16 steps of WMMA 16x16x4.
// ---------------------------------------------------------------------------
__global__ void __launch_bounds__(256) message_k(const float* __restrict__ E,
                                                 const float* __restrict__ Wm,
                                                 const float* __restrict__ T1,
                                                 const float* __restrict__ T2,
                                                 float* __restrict__ M) {
    __shared__ float sW[64 * 68];  // stride-68 pad: bank = (4r + c) % 64
    for (int idx = threadIdx.x; idx < 64 * 64; idx += 256) {
        int r = idx >> 6, c = idx & 63;
        sW[r * 68 + c] = Wm[r * 192 + 128 + c];  // Wm3 = Wm[:,128:192]
    }
    __syncthreads();

    const int lane = threadIdx.x & 31, lo = lane & 15, hi = lane >> 4;
    const int wave = (blockIdx.x << 3) | (threadIdx.x >> 5);
    const v8f vzero = {0.f, 0.f, 0.f, 0.f, 0.f, 0.f, 0.f, 0.f};

    for (int tile = wave; tile < B_ * N_ * (N_ / 32); tile += 1024 * 8) {
        int b = tile >> 11, rem = tile & 2047;   // 256*8 tiles per batch
        int i = rem >> 3, w0 = (rem & 7) << 5;   // 32-wide column tile

        v8f acc[8];
#pragma unroll
        for (int t = 0; t < 8; ++t) acc[t] = vzero;

        const float* Eb = E + (size_t)b * CE_ * NN_ + (size_t)i * N_ + w0 + lo;
#pragma unroll 4
        for (int c0 = 0; c0 < 64; c0 += 4) {
            int ca = c0 + 2 * hi;  // B frag V0: K={c0|c0+2}, V1: K={c0+1|c0+3}
            v2f bfA, bfB;
            bfA.x = Eb[(size_t)ca * NN_];
            bfA.y = Eb[(size_t)(ca + 1) * NN_];
            bfB.x = Eb[(size_t)ca * NN_ + 16];
            bfB.y = Eb[(size_t)(ca + 1) * NN_ + 16];
#pragma unroll
            for (int t = 0; t < 4; ++t) {
                v2f af = *(const v2f*)&sW[(t * 16 + lo) * 68 + ca];
                acc[t] = wmma4(af, bfA, acc[t]);
                acc[4 + t] = wmma4(af, bfB, acc[4 + t]);
            }
        }
        // epilogue: out row m = 16*t + v + 8*hi, cols = w0+lo and w0+16+lo
        const float* t1p = T1 + (b * N_ + i) * 64;
        const float* t2pA = T2 + (b * N_ + w0 + lo) * 64;
        const float* t2pB = t2pA + 16 * 64;
        float* outA = M + ((size_t)((b * N_ + i) * N_ + w0 + lo) << 6);
        float* outB = outA + 16 * 64;
#pragma unroll
        for (int t = 0; t < 4; ++t) {
            int mb = t * 16 + 8 * hi;
            float4 u0 = *(const float4*)&t1p[mb], u1 = *(const float4*)&t1p[mb + 4];
            float4 a0 = *(const float4*)&t2pA[mb], a1 = *(const float4*)&t2pA[mb + 4];
            float4 b0 = *(const float4*)&t2pB[mb], b1 = *(const float4*)&t2pB[mb + 4];
            float4 oA0, oA1, oB0, oB1;
            oA0.x = acc[t][0] + u0.x + a0.x;  oA0.y = acc[t][1] + u0.y + a0.y;
            oA0.z = acc[t][2] + u0.z + a0.z;  oA0.w = acc[t][3] + u0.w + a0.w;
            oA1.x = acc[t][4] + u1.x + a1.x;  oA1.y = acc[t][5] + u1.y + a1.y;
            oA1.z = acc[t][6] + u1.z + a1.z;  oA1.w = acc[t][7] + u1.w + a1.w;
            oB0.x = acc[4 + t][0] + u0.x + b0.x;  oB0.y = acc[4 + t][1] + u0.y + b0.y;
            oB0.z = acc[4 + t][2] + u0.z + b0.z;  oB0.w = acc[4 + t][3] + u0.w + b0.w;
            oB1.x = acc[4 + t][4] + u1.x + b1.x;  oB1.y = acc[4 + t][5] + u1.y + b1.y;
            oB1.z = acc[4 + t][6] + u1.z + b1.z;  oB1.w = acc[4 + t][7] + u1.w + b1.w;
            *(float4*)&outA[mb] = oA0;
            *(float4*)&outA[mb + 4] = oA1;
            *(float4*)&outB[mb] = oB0;
            *(float4*)&outB[mb + 4] = oB1;
        }
    }
}

// ---------------------------------------------------------------------------
// Link round.  FIRST:  A[b,i,j] = f(E[b,:,i,j])
//              else:   A[b,i,j] = f(Aprev[b,j,i] * M[b,j,i,:])
// f(v) = sigmoid(Wo . h + bo), h = sig(o)*tanh(sig(i)*tanh(ch)), g = Wx@v + b.
// f-gate (rows 64..127) is dead -> only 192 gate rows -> 12 k-tiles.
// One wave per 32-j tile: two column tiles share every A fragment.
// ---------------------------------------------------------------------------
template <bool FIRST>
__global__ void __launch_bounds__(256) link_k(const float* __restrict__ src,
                                              const float* __restrict__ Aprev,
                                              const float* __restrict__ Wx,
                                              const float* __restrict__ bl,
                                              const float* __restrict__ Wo,
                                              const float* __restrict__ bo,
                                              float* __restrict__ Aout) {
    __shared__ float sWx[192 * 68];  // remapped rows: [i | o | c_hat]
    __shared__ float sB[192];
    __shared__ float sWo[64];
    for (int idx = threadIdx.x; idx < 192 * 64; idx += 256) {
        int r = idx >> 6, c = idx & 63;
        int gr = (r < 64) ? r : r + 64;  // skip dead f-gate rows 64..127
        sWx[r * 68 + c] = Wx[gr * 64 + c];
    }
    if (threadIdx.x < 192) {
        int r = threadIdx.x;
        sB[r] = bl[(r < 64) ? r : r + 64];
        if (r < 64) sWo[r] = Wo[r];
    }
    __syncthreads();
    const float bo_v = bo[0];

    const int lane = threadIdx.x & 31, lo = lane & 15, hi = lane >> 4;
    const int wave = (blockIdx.x << 3) | (threadIdx.x >> 5);
    const v8f vzero = {0.f, 0.f, 0.f, 0.f, 0.f, 0.f, 0.f, 0.f};

    for (int tile = wave; tile < 16384; tile += 1024 * 8) {
        int b = tile >> 11, rem = tile & 2047;
        int i = rem >> 3, j0 = (rem & 7) << 5;

        v8f acc[24];
#pragma unroll
        for (int t = 0; t < 24; ++t) acc[t] = vzero;

        const float* Eb = nullptr;
        const float* vpA = nullptr;
        const float* vpB = nullptr;
        float sA = 1.f, sBs = 1.f;
        if (FIRST) {
            Eb = src + (size_t)b * CE_ * NN_ + (size_t)i * N_ + j0 + lo;
        } else {
            int jA = j0 + lo;
            vpA = src + ((size_t)((b * N_ + jA) * N_ + i) << 6);  // M[b,j,i,:]
            vpB = vpA + (size_t)16 * N_ * 64;
            sA = Aprev[(b * N_ + jA) * N_ + i];
            sBs = Aprev[(b * N_ + jA + 16) * N_ + i];
        }

        for (int c0 = 0; c0 < 64; c0 += 4) {
            int ca = c0 + 2 * hi;
            v2f bfA, bfB;
            if (FIRST) {
                bfA.x = Eb[(size_t)ca * NN_];
                bfA.y = Eb[(size_t)(ca + 1) * NN_];
                bfB.x = Eb[(size_t)ca * NN_ + 16];
                bfB.y = Eb[(size_t)(ca + 1) * NN_ + 16];
            } else {
                v2f ta = *(const v2f*)&vpA[ca];
                v2f tb = *(const v2f*)&vpB[ca];
                bfA.x = ta.x * sA;
                bfA.y = ta.y * sA;
                bfB.x = tb.x * sBs;
                bfB.y = tb.y * sBs;
            }
#pragma unroll
            for (int t = 0; t < 12; ++t) {
                v2f af = *(const v2f*)&sWx[(t * 16 + lo) * 68 + ca];
                acc[t] = wmma4(af, bfA, acc[t]);
                acc[12 + t] = wmma4(af, bfB, acc[12 + t]);
            }
        }

        // gate nonlinearity + Wo reduction.  D row = v + 8*hi within tile.
        float pA = 0.f, pB = 0.f;
#pragma unroll
        for (int kt = 0; kt < 4; ++kt) {
#pragma unroll
            for (int v = 0; v < 8; ++v) {
                int r = kt * 16 + v + 8 * hi;  // channel within gate (0..63)
                float bi = sB[r], bg = sB[64 + r], bc = sB[128 + r], wo = sWo[r];
                float ccA = sigm(acc[kt][v] + bi) * ftanh(acc[8 + kt][v] + bc);
                pA += wo * (sigm(acc[4 + kt][v] + bg) * ftanh(ccA));
                float ccB = sigm(acc[12 + kt][v] + bi) * ftanh(acc[20 + kt][v] + bc);
                pB += wo * (sigm(acc[16 + kt][v] + bg) * ftanh(ccB));
            }
        }
        pA += __shfl_xor(pA, 16, 32);  // combine rows 0-7 / 8-15 halves
        pB += __shfl_xor(pB, 16, 32);
        float aA = sigm(pA + bo_v);
        float aB = sigm(pB + bo_v);
        if (lane < 16) {
            int base = (b * N_ + i) * N_ + j0;
            Aout[base + lo] = aA;
            Aout[base + 16 + lo] = aB;
        }
    }
}

// ---------------------------------------------------------------------------
// m_sum[b,i,m] = sum_w A3[b,i,w] * M[b,i,w,m].  grid = B*N, 64 threads.
// ---------------------------------------------------------------------------
__global__ void msum_k(const float* __restrict__ A3, const float* __restrict__ M,
                       float* __restrict__ ms) {
    int bn = blockIdx.x;
    __shared__ float sA[256];
    for (int idx = threadIdx.x; idx < 256; idx += 64) sA[idx] = A3[bn * 256 + idx];
    __syncthreads();
    int m = threadIdx.x;
    const float* Mp = M + ((size_t)bn << 14) + m;
    float s = 0.f;
    for (int w = 0; w < 256; ++w) s += sA[w] * Mp[(size_t)w << 6];
    ms[bn * 64 + m] = s;
}

// ---------------------------------------------------------------------------
// GRU (node0 vs rest) + readout + log_softmax into pred[b,n,12].
// ---------------------------------------------------------------------------
__global__ void gru_k(const float* __restrict__ ms, const float* __restrict__ NF,
                      const float* Wih0, const float* bih0, const float* Whh0,
                      const float* bhh0, const float* Wih1, const float* bih1,
                      const float* Whh1, const float* bhh1, const float* Wr0,
                      const float* br0, const float* Wr1, const float* br1,
                      float* __restrict__ pred) {
    int bn = blockIdx.x, b = bn >> 8, n = bn & 255;
    bool n0 = (n == 0);
    const float* Wih = n0 ? Wih0 : Wih1;
    const float* bih = n0 ? bih0 : bih1;
    const float* Whh = n0 ? Whh0 : Whh1;
    const float* bhh = n0 ? bhh0 : bhh1;
    const float* Wr = n0 ? Wr0 : Wr1;
    const float* br = n0 ? br0 : br1;
    int S = n0 ? S_CLS_ : A_CLS_;

    __shared__ float sm[64], sh[64], sp[64], sl[A_CLS_], lse[1];
    int d = threadIdx.x;
    sm[d] = ms[bn * 64 + d];
    sh[d] = NF[(b * 64 + d) * 256 + n];
    __syncthreads();

    float ir = bih[d], iz = bih[64 + d], inn = bih[128 + d];
    float hr = bhh[d], hz = bhh[64 + d], hn = bhh[128 + d];
    for (int k = 0; k < 64; ++k) {
        float mv = sm[k], hv = sh[k];
        ir += Wih[d * 64 + k] * mv;
        iz += Wih[(64 + d) * 64 + k] * mv;
        inn += Wih[(128 + d) * 64 + k] * mv;
        hr += Whh[d * 64 + k] * hv;
        hz += Whh[(64 + d) * 64 + k] * hv;
        hn += Whh[(128 + d) * 64 + k] * hv;
    }
    float r = sigm(ir + hr), z = sigm(iz + hz);
    float nn = ftanh(inn + r * hn);
    sp[d] = (1.f - z) * nn + z * sh[d];
    __syncthreads();

    if (d < S) {
        float lg = br[d];
        for (int k = 0; k < 64; ++k) lg += Wr[d * 64 + k] * sp[k];
        sl[d] = lg;
    }
    __syncthreads();
    if (d == 0) {
        float mx = sl[0];
        for (int c = 1; c < S; ++c) mx = fmaxf(mx, sl[c]);
        float se = 0.f;
        for (int c = 0; c < S; ++c) se += __expf(sl[c] - mx);
        lse[0] = mx + logf(se);
    }
    __syncthreads();
    if (d < A_CLS_) pred[bn * A_CLS_ + d] = (d < S) ? (sl[d] - lse[0]) : 0.f;
}

// ---------------------------------------------------------------------------
extern "C" void kernel_launch(void* const* d_in, const int* in_sizes, int n_in,
                              void* d_out, int out_size, void* d_ws, size_t ws_size,
                              hipStream_t stream) {
    const float* E = (const float*)d_in[0];
    const float* NF = (const float*)d_in[1];
    // d_in[2] adj_mat, d_in[3] node_labels: unused by the reference math
    const float* Wx = (const float*)d_in[4];
    const float* bl = (const float*)d_in[5];
    const float* Wo = (const float*)d_in[6];
    const float* bo = (const float*)d_in[7];
    const float* Wm = (const float*)d_in[8];
    const float* bm = (const float*)d_in[9];
    const float* Wih0 = (const float*)d_in[10];
    const float* bih0 = (const float*)d_in[11];
    const float* Whh0 = (const float*)d_in[12];
    const float* bhh0 = (const float*)d_in[13];
    const float* Wih1 = (const float*)d_in[14];
    const float* bih1 = (const float*)d_in[15];
    const float* Whh1 = (const float*)d_in[16];
    const float* bhh1 = (const float*)d_in[17];
    const float* Wr0 = (const float*)d_in[18];
    const float* br0 = (const float*)d_in[19];
    const float* Wr1 = (const float*)d_in[20];
    const float* br1 = (const float*)d_in[21];

    // Workspace layout (~140 MB; M stays L2-resident on 192MB L2)
    char* ws = (char*)d_ws;
    float* M = (float*)(ws);                     // 134,217,728 B
    float* T1 = (float*)(ws + 134217728u);       //     524,288 B
    float* T2 = (float*)(ws + 134742016u);       //     524,288 B
    float* A1 = (float*)(ws + 135266304u);       //   2,097,152 B
    float* A2 = (float*)(ws + 137363456u);       //   2,097,152 B
    float* msb = (float*)(ws + 139460608u);      //     524,288 B

    float* A3 = (float*)d_out;                   // A output [B,N,N]
    float* pred = (float*)d_out + (size_t)B_ * N_ * N_;

    precomp_T<<<2048, 64, 0, stream>>>(NF, Wm, bm, T1, T2);
    message_k<<<1024, 256, 0, stream>>>(E, Wm, T1, T2, M);
    link_k<true><<<1024, 256, 0, stream>>>(E, nullptr, Wx, bl, Wo, bo, A1);
    link_k<false><<<1024, 256, 0, stream>>>(M, A1, Wx, bl, Wo, bo, A2);
    link_k<false><<<1024, 256, 0, stream>>>(M, A2, Wx, bl, Wo, bo, A3);
    msum_k<<<2048, 64, 0, stream>>>(A3, M, msb);
    gru_k<<<2048, 64, 0, stream>>>(msb, NF, Wih0, bih0, Whh0, bhh0, Wih1, bih1,
                                   Whh1, bhh1, Wr0, br0, Wr1, br1, pred);
}